// GraphAttentionLayer_18021682774974
// MI455X (gfx1250) — compile-verified
//
#include <hip/hip_runtime.h>
#include <math.h>

#define B_ 32
#define N_ 1024
#define C_ 1024
#define H_ 8
#define D_ 128

typedef float v2f __attribute__((ext_vector_type(2)));
typedef float v8f __attribute__((ext_vector_type(8)));
typedef unsigned int v4u __attribute__((ext_vector_type(4)));
typedef int v4i_t __attribute__((ext_vector_type(4)));
typedef int v8i_t __attribute__((ext_vector_type(8)));

static __device__ __forceinline__ v8f wmma_f32x4(v2f a, v2f b, v8f c) {
  // D(16x16,f32) = A(16x4,f32) * B(4x16,f32) + C
  return __builtin_amdgcn_wmma_f32_16x16x4_f32(false, a, false, b, (short)0, c,
                                               false, false);
}

// ---------------------------------------------------------------------------
// Tensor Data Mover: DMA a 2D tile (tile1 rows x tile0 f32 elems, row stride
// stride0 elems) from global memory into LDS at byte offset lds_byte.
// pad_code: LDS padding interval code (4 => +1 DWORD every 32 DWORDs -> LDS
// row stride 33; 6 => +1 DWORD every 128 DWORDs -> LDS row stride 129).
// Descriptor layout per CDNA5 ISA ch.8 (D# groups 0/1; groups 2/3 zero, 2D).
// ---------------------------------------------------------------------------
static __device__ __forceinline__ void tdm_load_2d(uint32_t lds_byte,
                                                   const void* gptr,
                                                   uint32_t tile0,
                                                   uint32_t tile1,
                                                   uint64_t stride0,
                                                   uint32_t pad_code) {
  const uint64_t ga = (uint64_t)(uintptr_t)gptr;
  v4u g0;
  g0.x = 1u;                                     // count=1 (valid user D#)
  g0.y = lds_byte;                               // lds_addr (bytes)
  g0.z = (uint32_t)ga;                           // global_addr[31:0]
  g0.w = (uint32_t)((ga >> 32) & 0x1ffffffu)     // global_addr[56:32]
         | (2u << 30);                           // type = 2 ("image")
  v8i_t g1;
  g1[0] = (int)((2u << 16)                       // data_size = 4 bytes
                | (1u << 20)                     // pad_enable
                | (pad_code << 22));             // pad_interval; pad_amount=0 (1 DW)
  g1[1] = (int)((tile0 & 0xffffu) << 16);        // tensor_dim0[15:0] (== tile0)
  g1[2] = (int)(((tile0 >> 16) & 0xffffu) |      // tensor_dim0[31:16]
                ((tile1 & 0xffffu) << 16));      // tensor_dim1[15:0] (== tile1)
  g1[3] = (int)(((tile1 >> 16) & 0xffffu) |      // tensor_dim1[31:16]
                ((tile0 & 0xffffu) << 16));      // tile_dim0
  g1[4] = (int)(tile1 & 0xffffu);                // tile_dim1 ; tile_dim2 = 0
  g1[5] = (int)(uint32_t)(stride0 & 0xffffffffu);      // dim0_stride[31:0]
  g1[6] = (int)(uint32_t)((stride0 >> 32) & 0xffffu);  // dim0_stride[47:32]
  g1[7] = 0;                                     // dim1_stride unused (2D)
  v4i_t z4 = {0, 0, 0, 0};
#if __clang_major__ >= 23
  v8i_t z8 = {0, 0, 0, 0, 0, 0, 0, 0};
  __builtin_amdgcn_tensor_load_to_lds(g0, g1, z4, z4, z8, 0);
#else
  __builtin_amdgcn_tensor_load_to_lds(g0, g1, z4, z4, 0);
#endif
}

// ---------------------------------------------------------------------------
// Kernel 1: x = h @ W^T + bias, written head-permuted as x_perm[b][H][n][d]
// 64x128 output tile per WG; K streamed in 32-chunks via double-buffered TDM.
// LDS: buf0 {As 64x33 @0, Bs 128x33 @2112}, buf1 {As @6336, Bs @8448}.
// ---------------------------------------------------------------------------
__global__ __launch_bounds__(256) void k_proj(const float* __restrict__ h,
                                              const float* __restrict__ W,
                                              const float* __restrict__ bias,
                                              float* __restrict__ xp) {
  extern __shared__ float sm[];
  const uint32_t lb = __builtin_amdgcn_groupstaticsize();
  const int tid = threadIdx.x;
  const int wave = tid >> 5, lane = tid & 31;
  const int lm = lane & 15, lh = lane >> 4;
  const int row0 = blockIdx.x * 64;
  const int col0 = blockIdx.y * 128;
  const int m_off = 16 * (wave & 3);
  const int n_off = 64 * (wave >> 2);

  if (wave == 0) {
    tdm_load_2d(lb + 0u, h + (size_t)row0 * C_, 32, 64, C_, 4);
    tdm_load_2d(lb + 2112u * 4u, W + (size_t)col0 * C_, 32, 128, C_, 4);
  }
  v8f acc[4] = {};
  for (int kt = 0; kt < 32; ++kt) {
    if (wave == 0) __builtin_amdgcn_s_wait_tensorcnt(0);
    __syncthreads();
    if (wave == 0 && kt + 1 < 32) {
      const uint32_t nbo = ((kt + 1) & 1) ? 6336u * 4u : 0u;
      tdm_load_2d(lb + nbo, h + (size_t)row0 * C_ + (kt + 1) * 32, 32, 64, C_,
                  4);
      tdm_load_2d(lb + nbo + 2112u * 4u,
                  W + (size_t)col0 * C_ + (kt + 1) * 32, 32, 128, C_, 4);
    }
    const int bsel = (kt & 1) ? 6336 : 0;
    const float* As = sm + bsel;
    const float* Bs = sm + bsel + 2112;
#pragma unroll
    for (int k0 = 0; k0 < 32; k0 += 4) {
      const int ak = k0 + 2 * lh;
      v2f a;
      a.x = As[(m_off + lm) * 33 + ak];
      a.y = As[(m_off + lm) * 33 + ak + 1];
#pragma unroll
      for (int nt = 0; nt < 4; ++nt) {
        v2f b;
        b.x = Bs[(n_off + nt * 16 + lm) * 33 + ak];
        b.y = Bs[(n_off + nt * 16 + lm) * 33 + ak + 1];
        acc[nt] = wmma_f32x4(a, b, acc[nt]);
      }
    }
    __syncthreads();
  }
  for (int nt = 0; nt < 4; ++nt) {
    int col = col0 + n_off + nt * 16 + lm;
    int hh = col >> 7, dd = col & 127;
    float bv = bias[col];
#pragma unroll
    for (int r = 0; r < 8; ++r) {
      int row = row0 + m_off + r + 8 * lh;
      int bb = row >> 10, nn = row & 1023;
      xp[(((size_t)bb * H_ + hh) * N_ + nn) * D_ + dd] = acc[nt][r] + bv;
    }
  }
}

// ---------------------------------------------------------------------------
// Kernel 2: xm = x_perm @ M (per-head metric), rows flattened [b*H*n][128]
// Whole M (128x129) + x tile (64x129) staged via TDM; K=128 one pass.
// ---------------------------------------------------------------------------
__global__ __launch_bounds__(256) void k_xm(const float* __restrict__ xp,
                                            const float* __restrict__ M,
                                            float* __restrict__ xm) {
  extern __shared__ float sm[];
  const float* As = sm;              // 64 x 129
  const float* Ms = sm + 64 * 129;   // 128 x 129
  const uint32_t lb = __builtin_amdgcn_groupstaticsize();
  const int tid = threadIdx.x;
  const int wave = tid >> 5, lane = tid & 31;
  const int lm = lane & 15, lh = lane >> 4;
  const size_t row0 = (size_t)blockIdx.x * 64;

  if (wave == 0) {
    tdm_load_2d(lb + 0u, xp + row0 * D_, 128, 64, 128, 6);
    tdm_load_2d(lb + 64u * 129u * 4u, M, 128, 128, 128, 6);
    __builtin_amdgcn_s_wait_tensorcnt(0);
  }
  __syncthreads();

  const int m_off = 16 * (wave & 3);
  const int n_off = 64 * (wave >> 2);
  v8f acc[4] = {};
#pragma unroll 4
  for (int k0 = 0; k0 < 128; k0 += 4) {
    const int ak = k0 + 2 * lh;
    v2f a;
    a.x = As[(m_off + lm) * 129 + ak];
    a.y = As[(m_off + lm) * 129 + ak + 1];
#pragma unroll
    for (int nt = 0; nt < 4; ++nt) {
      v2f b;
      b.x = Ms[ak * 129 + n_off + nt * 16 + lm];
      b.y = Ms[(ak + 1) * 129 + n_off + nt * 16 + lm];
      acc[nt] = wmma_f32x4(a, b, acc[nt]);
    }
  }
  for (int nt = 0; nt < 4; ++nt) {
    int e = n_off + nt * 16 + lm;
#pragma unroll
    for (int r = 0; r < 8; ++r) {
      size_t row = row0 + m_off + r + 8 * lh;
      xm[row * D_ + e] = acc[nt][r];
    }
  }
}

// ---------------------------------------------------------------------------
// Kernel 3: fused attention per (b*H, 32-query tile).
//   S[32 x 1025] strip lives in LDS, softmax in LDS, K/V (= x) streamed via
//   double-buffered TDM 64-row tiles; pre-LN result -> out[b][n][c].
// LDS floats: S @0 (32x1025), xq @32800 (32x129), xk0 @36928, xk1 @45184
// (each 64x129), red @53440 (256), rowv @53696 (32).
// ---------------------------------------------------------------------------
#define SOFF_XQ (32 * 1025)
#define SOFF_XK0 (SOFF_XQ + 32 * 129)
#define SOFF_XK1 (SOFF_XK0 + 64 * 129)
#define SOFF_RED (SOFF_XK1 + 64 * 129)
#define SOFF_ROWV (SOFF_RED + 256)

__global__ __launch_bounds__(256) void k_attn(const float* __restrict__ xp,
                                              const float* __restrict__ xm,
                                              float* __restrict__ out) {
  extern __shared__ float sm[];
  float* Ssm = sm;
  const float* xq = sm + SOFF_XQ;
  float* red = sm + SOFF_RED;
  float* rowv = sm + SOFF_ROWV;
  const uint32_t lb = __builtin_amdgcn_groupstaticsize();
  const int tid = threadIdx.x;
  const int wave = tid >> 5, lane = tid & 31;
  const int lm = lane & 15, lh = lane >> 4;
  const int bh = blockIdx.x;
  const int q0 = blockIdx.y * 32;
  const float* xk_base = xp + (size_t)bh * N_ * D_;
  const float* xq_base = xm + (size_t)bh * N_ * D_ + (size_t)q0 * D_;

  // ---- Phase 1: S = xMq @ xk^T over all 1024 keys ----
  const int qi = wave >> 2;  // 0..1 : query half
  const int ki = wave & 3;   // 0..3 : key quarter within 64-chunk
  if (wave == 0) {
    tdm_load_2d(lb + (uint32_t)SOFF_XQ * 4u, xq_base, 128, 32, 128, 6);
    tdm_load_2d(lb + (uint32_t)SOFF_XK0 * 4u, xk_base, 128, 64, 128, 6);
  }
  for (int kc = 0; kc < 16; ++kc) {
    if (wave == 0) __builtin_amdgcn_s_wait_tensorcnt(0);
    __syncthreads();
    if (wave == 0 && kc + 1 < 16) {
      const uint32_t nbo =
          ((kc + 1) & 1) ? (uint32_t)SOFF_XK1 * 4u : (uint32_t)SOFF_XK0 * 4u;
      tdm_load_2d(lb + nbo, xk_base + (size_t)(kc + 1) * 64 * D_, 128, 64, 128,
                  6);
    }
    const float* xk = sm + ((kc & 1) ? SOFF_XK1 : SOFF_XK0);
    v8f acc = {};
#pragma unroll 4
    for (int k0 = 0; k0 < 128; k0 += 4) {
      const int ak = k0 + 2 * lh;
      v2f a;
      a.x = xq[(qi * 16 + lm) * 129 + ak];
      a.y = xq[(qi * 16 + lm) * 129 + ak + 1];
      v2f b;
      b.x = xk[(ki * 16 + lm) * 129 + ak];
      b.y = xk[(ki * 16 + lm) * 129 + ak + 1];
      acc = wmma_f32x4(a, b, acc);
    }
#pragma unroll
    for (int r = 0; r < 8; ++r)
      Ssm[(qi * 16 + r + 8 * lh) * 1025 + kc * 64 + ki * 16 + lm] = acc[r];
    __syncthreads();
  }

  // ---- Phase 2: softmax over each row of S (8 threads per row) ----
  {
    const int q = tid >> 3, seg = tid & 7;
    float* Srow = Ssm + q * 1025 + seg * 128;
    float mx = -3.402823466e38f;
    for (int i = 0; i < 128; ++i) mx = fmaxf(mx, Srow[i]);
    red[tid] = mx;
    __syncthreads();
    if (seg == 0) {
      float m2 = red[q * 8];
      for (int j = 1; j < 8; ++j) m2 = fmaxf(m2, red[q * 8 + j]);
      rowv[q] = m2;
    }
    __syncthreads();
    const float m2 = rowv[q];
    float s = 0.f;
    for (int i = 0; i < 128; ++i) {
      float e = expf(Srow[i] - m2);
      Srow[i] = e;
      s += e;
    }
    red[tid] = s;
    __syncthreads();
    if (seg == 0) {
      float ss = 0.f;
      for (int j = 0; j < 8; ++j) ss += red[q * 8 + j];
      rowv[q] = 1.f / ss;
    }
    __syncthreads();
    const float inv = rowv[q];
    for (int i = 0; i < 128; ++i) Srow[i] *= inv;
  }

  // ---- Phase 3: O = P @ x (K = 1024 keys, double-buffered TDM stream;
  //      first DMA overlaps the tail of softmax) ----
  const int eq = wave & 3;  // e-column group
  if (wave == 0)
    tdm_load_2d(lb + (uint32_t)SOFF_XK0 * 4u, xk_base, 128, 64, 128, 6);
  v8f oacc[2] = {};
  for (int kc = 0; kc < 16; ++kc) {
    if (wave == 0) __builtin_amdgcn_s_wait_tensorcnt(0);
    __syncthreads();
    if (wave == 0 && kc + 1 < 16) {
      const uint32_t nbo =
          ((kc + 1) & 1) ? (uint32_t)SOFF_XK1 * 4u : (uint32_t)SOFF_XK0 * 4u;
      tdm_load_2d(lb + nbo, xk_base + (size_t)(kc + 1) * 64 * D_, 128, 64, 128,
                  6);
    }
    const float* xk = sm + ((kc & 1) ? SOFF_XK1 : SOFF_XK0);
#pragma unroll 4
    for (int k0 = 0; k0 < 64; k0 += 4) {
      const int ak = k0 + 2 * lh;
      v2f a;
      a.x = Ssm[(qi * 16 + lm) * 1025 + kc * 64 + ak];
      a.y = Ssm[(qi * 16 + lm) * 1025 + kc * 64 + ak + 1];
#pragma unroll
      for (int j = 0; j < 2; ++j) {
        const int e_off = eq * 32 + j * 16;
        v2f b;
        b.x = xk[ak * 129 + e_off + lm];
        b.y = xk[(ak + 1) * 129 + e_off + lm];
        oacc[j] = wmma_f32x4(a, b, oacc[j]);
      }
    }
    __syncthreads();
  }

  // ---- Phase 4: store pre-LN output into out[b][n][c] ----
  const int b_idx = bh >> 3, h_idx = bh & 7;
  for (int j = 0; j < 2; ++j) {
    const int e_off = eq * 32 + j * 16;
    const int cidx = h_idx * D_ + e_off + lm;
#pragma unroll
    for (int r = 0; r < 8; ++r) {
      int n = q0 + qi * 16 + r + 8 * lh;
      out[((size_t)b_idx * N_ + n) * C_ + cidx] = oacc[j][r];
    }
  }
}

// ---------------------------------------------------------------------------
// Kernel 4: in-place LayerNorm + exact-erf GELU, one WG per row of 1024.
// ---------------------------------------------------------------------------
__global__ __launch_bounds__(256) void k_lngelu(float* __restrict__ out,
                                                const float* __restrict__ gamma,
                                                const float* __restrict__ beta) {
  __shared__ float red[256];
  const size_t row = blockIdx.x;
  float* p = out + row * C_;
  const int tid = threadIdx.x;
  float v[4];
  float s = 0.f;
#pragma unroll
  for (int i = 0; i < 4; ++i) {
    v[i] = p[tid + 256 * i];
    s += v[i];
  }
  red[tid] = s;
  __syncthreads();
  for (int off = 128; off > 0; off >>= 1) {
    if (tid < off) red[tid] += red[tid + off];
    __syncthreads();
  }
  const float mu = red[0] * (1.0f / C_);
  __syncthreads();
  float sv = 0.f;
#pragma unroll
  for (int i = 0; i < 4; ++i) {
    float d = v[i] - mu;
    sv += d * d;
  }
  red[tid] = sv;
  __syncthreads();
  for (int off = 128; off > 0; off >>= 1) {
    if (tid < off) red[tid] += red[tid + off];
    __syncthreads();
  }
  const float rs = rsqrtf(red[0] * (1.0f / C_) + 1e-5f);
#pragma unroll
  for (int i = 0; i < 4; ++i) {
    int c = tid + 256 * i;
    float y = (v[i] - mu) * rs * gamma[c] + beta[c];
    p[c] = 0.5f * y * (1.f + erff(y * 0.70710678118654752f));
  }
}

// ---------------------------------------------------------------------------
extern "C" void kernel_launch(void* const* d_in, const int* in_sizes, int n_in,
                              void* d_out, int out_size, void* d_ws,
                              size_t ws_size, hipStream_t stream) {
  const float* h = (const float*)d_in[0];
  const float* W = (const float*)d_in[1];
  const float* Wb = (const float*)d_in[2];
  const float* M = (const float*)d_in[3];
  const float* g = (const float*)d_in[4];
  const float* be = (const float*)d_in[5];
  float* out = (float*)d_out;

  float* xp = (float*)d_ws;                    // [b*H*n][d] = 128 MB
  float* xm = xp + (size_t)B_ * H_ * N_ * D_;  // [b*H*n][d] = 128 MB

  const size_t smA = 2 * (64 * 33 + 128 * 33) * sizeof(float);  // ~50 KB
  const size_t smB = (64 * 129 + 128 * 129) * sizeof(float);    // ~97 KB
  const size_t smC = (SOFF_ROWV + 32) * sizeof(float);          // ~215 KB

  (void)hipFuncSetAttribute(reinterpret_cast<const void*>(k_xm),
                            hipFuncAttributeMaxDynamicSharedMemorySize,
                            (int)smB);
  (void)hipFuncSetAttribute(reinterpret_cast<const void*>(k_attn),
                            hipFuncAttributeMaxDynamicSharedMemorySize,
                            (int)smC);

  hipLaunchKernelGGL(k_proj, dim3((B_ * N_) / 64, C_ / 128), dim3(256), smA,
                     stream, h, W, Wb, xp);
  hipLaunchKernelGGL(k_xm, dim3((B_ * H_ * N_) / 64), dim3(256), smB, stream,
                     xp, M, xm);
  hipLaunchKernelGGL(k_attn, dim3(B_ * H_, N_ / 32), dim3(256), smC, stream,
                     xp, xm, out);
  hipLaunchKernelGGL(k_lngelu, dim3(B_ * N_), dim3(256), 0, stream, out, g, be);
}